// NeighborEmbedding_79508434583953
// MI455X (gfx1250) — compile-verified
//
#include <hip/hip_runtime.h>
#include <math.h>

#define N_NODES 50000
#define N_EDGES 1600000
#define NUM_RBF 32
#define NUM_CH 64
#define NUM_SPECIES 100
#define CUTOFF 5.0f

typedef float v2f __attribute__((ext_vector_type(2)));
typedef float v8f __attribute__((ext_vector_type(8)));

#define N_TILES_EDGE (N_EDGES / 16)   // 100000, exact
#define N_TILES_NODE (N_NODES / 16)   // 3125, exact

#define EDGE_BLOCKS 1250
#define EDGE_WAVES  (EDGE_BLOCKS * 8) // 10000 -> exactly 10 tiles per wave
#define NODE_BLOCKS 98
#define NODE_WAVES  (NODE_BLOCKS * 8) // 784

// ---------------------------------------------------------------------------
// Kernel 0: zero the aggregation buffer (d_out doubles as agg scratch).
// ---------------------------------------------------------------------------
__global__ __launch_bounds__(256) void zero_f4_kernel(float4* __restrict__ p, int n4) {
    int i = blockIdx.x * 256 + threadIdx.x;
    if (i < n4) p[i] = make_float4(0.f, 0.f, 0.f, 0.f);
}

// ---------------------------------------------------------------------------
// Kernel 1: persistent waves, 16-edge tiles.
//   W[16,64] = edge_feats[16,32] @ dp_kernel[32,64] + dp_bias (in acc init)
//   msg      = W * cutoff(r)[row] * emb[z[senders[row]]][col]
//   atomic add into agg[receivers[row]][col]
// B fragments (dp_kernel) are tile-invariant -> hoisted into 64 VGPRs/lane.
// A frag (16x4 f32): lane<16 -> M=lane, K=k0+{0,1}; lane>=16 -> M=lane-16, K=k0+{2,3}
// B frag (4x16 f32): lane<16 -> K=k0+{0,1}, N=lane; lane>=16 -> K=k0+{2,3}, N=lane-16
// C/D  (16x16 f32): VGPR i -> M = i + 8*(lane>=16), N = lane%16
// ---------------------------------------------------------------------------
__global__ __launch_bounds__(256) void edge_msg_kernel(
    const float* __restrict__ edge_feats,   // [E,32]
    const float* __restrict__ edge_weight,  // [E]
    const int*   __restrict__ senders,      // [E]
    const int*   __restrict__ receivers,    // [E]
    const int*   __restrict__ node_z,       // [N]
    const float* __restrict__ emb_table,    // [100,64]
    const float* __restrict__ dp_kernel,    // [32,64]
    const float* __restrict__ dp_bias,      // [64]
    float*       __restrict__ agg)          // [N,64] (pre-zeroed)
{
    __shared__ float s_emb[NUM_SPECIES * NUM_CH];   // 25.6 KB

    const int t = threadIdx.x;
    for (int i = t; i < NUM_SPECIES * NUM_CH; i += 256) s_emb[i] = emb_table[i];
    __syncthreads();

    const int lane  = t & 31;
    const int row   = lane & 15;
    const int half  = lane >> 4;
    const int gwave = blockIdx.x * 8 + (t >> 5);    // 0..EDGE_WAVES-1

    // Tile-invariant B fragments + bias: load once, keep in registers.
    v2f breg[NUM_RBF / 4][4];
#pragma unroll
    for (int kk = 0; kk < NUM_RBF / 4; ++kk) {
        const int k = kk * 4 + half * 2;
#pragma unroll
        for (int nt = 0; nt < 4; ++nt) {
            const int col = nt * 16 + row;
            v2f b;
            b.x = dp_kernel[(k)     * NUM_CH + col];
            b.y = dp_kernel[(k + 1) * NUM_CH + col];
            breg[kk][nt] = b;
        }
    }
    // Bias splat per N-tile: bias[col] is lane-constant across the 8 acc rows,
    // so it can seed the WMMA accumulator (saves 32 v_add per tile).
    v8f bias_splat[4];
#pragma unroll
    for (int nt = 0; nt < 4; ++nt) {
        const float b = dp_bias[nt * 16 + row];
#pragma unroll
        for (int j = 0; j < 8; ++j) bias_splat[nt][j] = b;
    }

    // 100000 tiles / 10000 waves == 10 iterations for every wave: no divergence,
    // EXEC stays all-ones around the WMMAs.
    for (int tile = gwave; tile < N_TILES_EDGE; tile += EDGE_WAVES) {
        const int e0 = tile * 16;

        // Prefetch the next tile's A data (2KB, 64B per lane) into cache while
        // this tile's WMMA + atomic work covers the HBM latency. Wave-uniform
        // guard -> no EXEC divergence around the WMMAs.
        if (tile + EDGE_WAVES < N_TILES_EDGE) {
            const char* pf = (const char*)(edge_feats
                              + (size_t)(e0 + EDGE_WAVES * 16) * NUM_RBF) + lane * 64;
            __builtin_prefetch(pf, 0, 1);
        }

        // Per-row metadata (both lane halves compute row = lane&15 identically).
        const float r   = edge_weight[e0 + row];
        const float cut = (r < CUTOFF) ? 0.5f * (cosf(r * 0.62831853071795864769f) + 1.0f)
                                       : 0.0f;
        const int   snd = senders[e0 + row];
        const int   rcv = receivers[e0 + row];
        const int   z   = node_z[snd];

        const float* arow = edge_feats + (size_t)(e0 + row) * NUM_RBF;

        v8f acc[4];
#pragma unroll
        for (int nt = 0; nt < 4; ++nt) acc[nt] = bias_splat[nt];

#pragma unroll
        for (int kk = 0; kk < NUM_RBF / 4; ++kk) {
            const int k = kk * 4 + half * 2;
            v2f a;
            a.x = arow[k];
            a.y = arow[k + 1];
#pragma unroll
            for (int nt = 0; nt < 4; ++nt) {
                acc[nt] = __builtin_amdgcn_wmma_f32_16x16x4_f32(
                    false, a, false, breg[kk][nt], (short)0, acc[nt], false, false);
            }
        }

        // Epilogue: cutoff + emb gather (LDS) + atomic scatter-add (L2-resident).
#pragma unroll
        for (int i = 0; i < 8; ++i) {
            const int   m     = i + half * 8;       // edge row of this acc element
            const float cut_m = __shfl(cut, m);
            const int   z_m   = __shfl(z,   m);
            const int   rcv_m = __shfl(rcv, m);
#pragma unroll
            for (int nt = 0; nt < 4; ++nt) {
                const int   col = nt * 16 + row;
                const float v   = acc[nt][i] * cut_m * s_emb[z_m * NUM_CH + col];
                unsafeAtomicAdd(&agg[(size_t)rcv_m * NUM_CH + col], v);
            }
        }
    }
}

// ---------------------------------------------------------------------------
// Kernel 2: out = concat(node_feats, agg) @ comb_kernel + comb_bias
// comb_kernel staged in LDS with K-paired float2 layout so each B fragment is
// one contiguous ds_load_b64; comb_bias seeds the accumulators.
// agg lives in `out`; each wave reads only its own 16 rows, then overwrites.
// ---------------------------------------------------------------------------
__global__ __launch_bounds__(256) void node_comb_kernel(
    const float* __restrict__ node_feats,   // [N,64]
    const float* __restrict__ comb_kernel,  // [128,64]
    const float* __restrict__ comb_bias,    // [64]
    float*       __restrict__ out)          // [N,64] in: agg, out: result
{
    // s_ck2[(k/2)*64 + col] = { ck[k][col], ck[k+1][col] }  (k even)
    __shared__ v2f s_ck2[NUM_CH * NUM_CH];   // 64 kpairs * 64 cols * 8B = 32 KB

    const int t = threadIdx.x;
    for (int i = t; i < 2 * NUM_CH * NUM_CH; i += 256) {
        const int k   = i / NUM_CH;            // coalesced global read
        const int col = i % NUM_CH;
        ((float*)&s_ck2[(k >> 1) * NUM_CH + col])[k & 1] = comb_kernel[i];
    }
    __syncthreads();

    const int lane  = t & 31;
    const int row   = lane & 15;
    const int half  = lane >> 4;
    const int gwave = blockIdx.x * 8 + (t >> 5);

    v8f bias_splat[4];
#pragma unroll
    for (int nt = 0; nt < 4; ++nt) {
        const float b = comb_bias[nt * 16 + row];
#pragma unroll
        for (int j = 0; j < 8; ++j) bias_splat[nt][j] = b;
    }

    for (int tile = gwave; tile < N_TILES_NODE; tile += NODE_WAVES) {
        const int n0 = tile * 16;

        const float* nf = node_feats + (size_t)(n0 + row) * NUM_CH;
        const float* ag = out        + (size_t)(n0 + row) * NUM_CH;

        v8f acc[4];
#pragma unroll
        for (int nt = 0; nt < 4; ++nt) acc[nt] = bias_splat[nt];

#pragma unroll
        for (int kk = 0; kk < (2 * NUM_CH) / 4; ++kk) {
            const int k  = kk * 4 + half * 2;   // chunk never straddles the 64 boundary
            const int kp = kk * 2 + half;       // k / 2
            v2f a;
            if (k < NUM_CH) { a.x = nf[k];          a.y = nf[k + 1]; }
            else            { a.x = ag[k - NUM_CH]; a.y = ag[k - NUM_CH + 1]; }
#pragma unroll
            for (int nt = 0; nt < 4; ++nt) {
                const int col = nt * 16 + row;
                const v2f b   = s_ck2[kp * NUM_CH + col];   // single ds_load_b64
                acc[nt] = __builtin_amdgcn_wmma_f32_16x16x4_f32(
                    false, a, false, b, (short)0, acc[nt], false, false);
            }
        }

        // All agg loads already consumed by WMMA, but make the load->store
        // ordering on `out` explicit before overwriting the rows we read.
        asm volatile("s_wait_loadcnt 0" ::: "memory");

#pragma unroll
        for (int i = 0; i < 8; ++i) {
            const int m = i + half * 8;
            float* orow = out + (size_t)(n0 + m) * NUM_CH;
#pragma unroll
            for (int nt = 0; nt < 4; ++nt) {
                const int col = nt * 16 + row;
                orow[col] = acc[nt][i];
            }
        }
    }
}

// ---------------------------------------------------------------------------
extern "C" void kernel_launch(void* const* d_in, const int* in_sizes, int n_in,
                              void* d_out, int out_size, void* d_ws, size_t ws_size,
                              hipStream_t stream) {
    const int*   node_z      = (const int*)  d_in[0];
    const float* node_feats  = (const float*)d_in[1];
    const int*   senders     = (const int*)  d_in[2];
    const int*   receivers   = (const int*)  d_in[3];
    const float* edge_weight = (const float*)d_in[4];
    const float* edge_feats  = (const float*)d_in[5];
    const float* emb_table   = (const float*)d_in[6];
    const float* dp_kernel   = (const float*)d_in[7];
    const float* dp_bias     = (const float*)d_in[8];
    const float* comb_kernel = (const float*)d_in[9];
    const float* comb_bias   = (const float*)d_in[10];
    float* out = (float*)d_out;

    // 1) zero aggregation buffer (d_out): 50000*64 floats = 800000 float4
    const int n4 = (N_NODES * NUM_CH) / 4;
    zero_f4_kernel<<<(n4 + 255) / 256, 256, 0, stream>>>((float4*)out, n4);

    // 2) edge messages + scatter-add into d_out (persistent waves)
    edge_msg_kernel<<<EDGE_BLOCKS, 256, 0, stream>>>(
        edge_feats, edge_weight, senders, receivers, node_z,
        emb_table, dp_kernel, dp_bias, out);

    // 3) node combine GEMM (reads agg rows from d_out, overwrites with result)
    node_comb_kernel<<<NODE_BLOCKS, 256, 0, stream>>>(
        node_feats, comb_kernel, comb_bias, out);
}